// RotaryEmbedding_6717328850968
// MI455X (gfx1250) — compile-verified
//
#include <hip/hip_runtime.h>
#include <math.h>

// Problem constants (fixed by the reference): x is (B=8, T=8192, D=1024) fp32.
#define B_DIM 8
#define T_DIM 8192
#define D_DIM 1024
#define H_DIM 512
// LDS row padding: row stride 1032 floats -> stride % 64 banks == 8, which makes
// the batch-interleaved gather (bb in {0,2,4,6} / {1,3,5,7} at same 2k) conflict-free.
#define ROWPAD 1032

// Native clang vector type so __builtin_nontemporal_store accepts it
// (HIP's float4 is a HIP_vector_type class, which the builtin rejects).
typedef float v4f __attribute__((ext_vector_type(4)));

__global__ __launch_bounds__(256)
void RotaryEmbedding_kernel(const float* __restrict__ x, float* __restrict__ out) {
    __shared__ float xs[B_DIM * ROWPAD];   // x[:, m, :] staged, padded rows
    __shared__ float cs[H_DIM];            // cos(m * theta_j)
    __shared__ float sn[H_DIM];            // sin(m * theta_j)

    const int tid = threadIdx.x;
    const int m   = blockIdx.x;            // timestep handled by this block

    // ---- Stage x[:, m, :] (8 rows x 1024 floats = 32KB) into LDS via the
    // ---- CDNA5 async global->LDS path (tracked with ASYNCcnt). Each thread
    // ---- issues one b128 per batch row; per row, lane tid loads floats
    // ---- [4*tid .. 4*tid+3] -> perfectly coalesced. Tiles are single-use
    // ---- (all reuse is in LDS), so mark non-temporal to spare L2.
    const float* gbase = x + (size_t)m * D_DIM;
    #pragma unroll
    for (int i = 0; i < B_DIM; ++i) {
        unsigned long long gaddr =
            (unsigned long long)(gbase + (size_t)i * (size_t)T_DIM * D_DIM + (size_t)(tid << 2));
        // Generic pointers to LDS carry the LDS byte offset in their low 32 bits.
        unsigned lds_off = (unsigned)(unsigned long long)(&xs[i * ROWPAD + (tid << 2)]);
        asm volatile("global_load_async_to_lds_b128 %0, %1, off th:TH_LOAD_NT"
                     :: "v"(lds_off), "v"(gaddr)
                     : "memory");
    }

    // ---- Overlap: compute the 512 sin/cos values while the async DMA flies.
    // theta_j = 10000^(-2j/1024) = exp2(-2j/1024 * log2(10000))
    const float mF = (float)m;
    for (int j = tid; j < H_DIM; j += 256) {
        float theta = exp2f((float)j * (-2.0f / (float)D_DIM) * 13.287712379549449f);
        float ang = mF * theta;
        float s, c;
        sincosf(ang, &s, &c);   // accurate range reduction (angles up to ~8191 rad)
        cs[j] = c;
        sn[j] = s;
    }

    // Wait for this wave's async LDS writes, then sync the workgroup so all
    // waves see the full 32KB tile + the sin/cos tables.
    asm volatile("s_wait_asynccnt 0" ::: "memory");
    __syncthreads();

    // ---- Produce 8 float4 outputs per thread. For chunk i, flat float4 index
    // q = tid + 256*i spans exactly batch-row i, col4 = tid, and the flat
    // element index is linear = 4q + e, giving closed-form gather indices:
    //   k      = q >> 2
    //   bb(e)  = (2q + (e>>1)) & 7
    //   c(e)   = e & 1     (c==0 -> -x[bb,2k+1], c==1 -> +x[bb,2k])
    #pragma unroll
    for (int i = 0; i < B_DIM; ++i) {
        const int q = tid + (i << 8);
        const int lbase = i * ROWPAD + (tid << 2);
        const v4f xv = *(const v4f*)&xs[lbase];   // ds_load_b128

        const int k2  = (q >> 2) << 1;           // 2k
        const int bb0 = (2 * q) & 7;             // for e = 0,1
        const int bb1 = (2 * q + 1) & 7;         // for e = 2,3
        const float* p0 = &xs[bb0 * ROWPAD + k2];
        const float* p1 = &xs[bb1 * ROWPAD + k2];
        const float a0 = p0[0], a1 = p0[1];
        const float b0 = p1[0], b1 = p1[1];

        const int j0 = tid << 1;                 // col>>1 for e = 0,1
        const float c0 = cs[j0],     s0 = sn[j0];
        const float c1 = cs[j0 + 1], s1 = sn[j0 + 1];

        v4f o;
        o.x = fmaf(s0, -a1, c0 * xv.x);          // e=0: rot = -x[bb0, 2k+1]
        o.y = fmaf(s0,  a0, c0 * xv.y);          // e=1: rot = +x[bb0, 2k]
        o.z = fmaf(s1, -b1, c1 * xv.z);          // e=2: rot = -x[bb1, 2k+1]
        o.w = fmaf(s1,  b0, c1 * xv.w);          // e=3: rot = +x[bb1, 2k]

        float* op = out + ((size_t)i * T_DIM + (size_t)m) * D_DIM + (size_t)(tid << 2);
        // Streaming output: write once, never re-read -> non-temporal b128 store.
        __builtin_nontemporal_store(o, (v4f*)op);
    }
}

extern "C" void kernel_launch(void* const* d_in, const int* in_sizes, int n_in,
                              void* d_out, int out_size, void* d_ws, size_t ws_size,
                              hipStream_t stream) {
    (void)in_sizes; (void)n_in; (void)d_ws; (void)ws_size; (void)out_size;
    const float* x = (const float*)d_in[0];
    float* out = (float*)d_out;
    // One block per timestep; 256 threads (8 wave32) per block.
    RotaryEmbedding_kernel<<<dim3(T_DIM), dim3(256), 0, stream>>>(x, out);
}